// GNN_maker_HNN_16844861735803
// MI455X (gfx1250) — compile-verified
//
#include <hip/hip_runtime.h>
#include <cmath>

// ---------------------------------------------------------------------------
// GNN pipeline for MI455X (gfx1250, wave32).
//   per layer: m = h @ W^T + b  (WMMA f32 16x16x4, full precision)
//              agg[dst] += m[src] over 1.6M edges (L2-resident f32 atomics)
//   tanh fused into the next layer's A-fragment load.
// Memory-bound: ~2.5 GB of mostly-L2 traffic vs 2.5 GFLOP of math.
// ---------------------------------------------------------------------------

typedef float v2f __attribute__((ext_vector_type(2)));
typedef float v8f __attribute__((ext_vector_type(8)));

#define D 64
#define TILES_PER_BLOCK 4                 // waves per block (1 16-row tile per wave)
#define LIN_BLOCK (TILES_PER_BLOCK * 32)

// ---------------------------------------------------------------------------
// Dense per-node linear: out[n, j] = sum_k act(h[n,k]) * W[j,k] + b[j]
// One wave = one 16-row tile, all 64 output columns (4 N-tiles of 16).
// A fragment layout (f32 16x16x4, ISA 7.12.2): lane L holds row (L&15),
// K = 4*ks + 2*(L>>4) + {0,1}.  B[k][j] = W[j][k] has the same lane pairing.
// n_nodes must be a multiple of 16 (100000 = 6250*16) so EXEC stays all-1s.
// ---------------------------------------------------------------------------
__global__ __launch_bounds__(LIN_BLOCK)
void linear_wmma_kernel(const float* __restrict__ h, const float* __restrict__ W,
                        const float* __restrict__ bias, float* __restrict__ out,
                        int n_nodes, int apply_tanh)
{
    __shared__ float sW[D * D];
    __shared__ float sB[D];
    for (int i = threadIdx.x; i < D * D; i += blockDim.x) sW[i] = W[i];
    if (threadIdx.x < D) sB[threadIdx.x] = bias[threadIdx.x];
    __syncthreads();

    const int wave    = threadIdx.x >> 5;
    const int lane    = threadIdx.x & 31;
    const int rowTile = blockIdx.x * TILES_PER_BLOCK + wave;
    if (rowTile * 16 >= n_nodes) return;   // wave-uniform guard -> EXEC all-1s

    const int mrow    = lane & 15;         // A-matrix row within tile
    const int kOff    = (lane >> 4) * 2;   // high half-wave holds K+2, K+3
    const int rowBase = rowTile * 16;

    // Preload all 16 A fragments (K=64 as 16 steps of K=4), fusing tanh.
    v2f a[16];
    const float* hp = h + (size_t)(rowBase + mrow) * D + kOff;
    #pragma unroll
    for (int ks = 0; ks < 16; ++ks) {
        float a0 = hp[ks * 4];
        float a1 = hp[ks * 4 + 1];
        if (apply_tanh) { a0 = tanhf(a0); a1 = tanhf(a1); }
        v2f t; t.x = a0; t.y = a1;
        a[ks] = t;
    }

    #pragma unroll
    for (int nt = 0; nt < 4; ++nt) {
        const int col = nt * 16 + (lane & 15);
        const float bval = sB[col];                 // bias depends only on N
        v8f c = {bval, bval, bval, bval, bval, bval, bval, bval};
        const float* wp = sW + col * D + kOff;      // B[k][col] = W[col][k]
        #pragma unroll
        for (int ks = 0; ks < 16; ++ks) {
            v2f bf; bf.x = wp[ks * 4]; bf.y = wp[ks * 4 + 1];
            // (neg_a, A, neg_b, B, c_mod, C, reuse_a, reuse_b)
            c = __builtin_amdgcn_wmma_f32_16x16x4_f32(false, a[ks], false, bf,
                                                      (short)0, c, false, false);
        }
        // D layout: VGPR v -> row v (lanes 0-15) / row v+8 (lanes 16-31)
        const int rHi = (lane >> 4) * 8;
        float* op = out + (size_t)(rowBase + rHi) * D + col;
        #pragma unroll
        for (int v = 0; v < 8; ++v)
            op[(size_t)v * D] = c[v];
    }
}

// ---------------------------------------------------------------------------
// Edge scatter: out[dst[e], :] += m[src[e], :].
// 32 lanes per edge, 2 features per lane -> 2 f32 atomics (no-return -> L2).
// Both m (gather) and out (atomics) are 25.6 MB -> resident in 192 MB L2.
// ---------------------------------------------------------------------------
__global__ __launch_bounds__(256)
void scatter_add_kernel(const float* __restrict__ m, const int* __restrict__ src,
                        const int* __restrict__ dst, float* __restrict__ out, int E)
{
    const unsigned gid = blockIdx.x * blockDim.x + threadIdx.x;
    const unsigned e = gid >> 5;
    if (e >= (unsigned)E) return;
    const int f  = gid & 31;
    const int s  = src[e];
    const int d0 = dst[e];
    const float* mp = m + (size_t)s * D + f;
    float* op = out + (size_t)d0 * D + f;
    atomicAdd(op,      mp[0]);
    atomicAdd(op + 32, mp[32]);
}

// ---------------------------------------------------------------------------
__global__ __launch_bounds__(256)
void zero_kernel(float* __restrict__ p, size_t n)
{
    const size_t i = (size_t)blockIdx.x * blockDim.x + threadIdx.x;
    if (i < n) p[i] = 0.0f;
}

// ---------------------------------------------------------------------------
// Final reduction: per_graph[g] += sum_f H[n, f] for n with node2graph[n]==g.
// One wave per node; lane-sums + shuffle tree; 8 LDS bins, 8 atomics/block.
// ---------------------------------------------------------------------------
__global__ __launch_bounds__(256)
void graph_reduce_kernel(const float* __restrict__ H, const int* __restrict__ n2g,
                         float* __restrict__ out, int n_nodes)
{
    __shared__ float bins[8];
    if (threadIdx.x < 8) bins[threadIdx.x] = 0.0f;
    __syncthreads();

    const int wavesPerBlock = blockDim.x >> 5;
    const int wave = threadIdx.x >> 5;
    const int lane = threadIdx.x & 31;
    for (int n = blockIdx.x * wavesPerBlock + wave; n < n_nodes;
         n += gridDim.x * wavesPerBlock) {
        const float* hp = H + (size_t)n * D;
        float s = hp[lane] + hp[lane + 32];
        #pragma unroll
        for (int off = 16; off > 0; off >>= 1)
            s += __shfl_down(s, off, 32);          // wave32 shuffle tree
        if (lane == 0) atomicAdd(&bins[n2g[n]], s);
    }
    __syncthreads();
    if (threadIdx.x < 8) atomicAdd(&out[threadIdx.x], bins[threadIdx.x]);
}

// ---------------------------------------------------------------------------
extern "C" void kernel_launch(void* const* d_in, const int* in_sizes, int n_in,
                              void* d_out, int out_size, void* d_ws, size_t ws_size,
                              hipStream_t stream)
{
    const float* x   = (const float*)d_in[0];
    const float* W1  = (const float*)d_in[1];
    const float* b1  = (const float*)d_in[2];
    const float* W2  = (const float*)d_in[3];
    const float* b2  = (const float*)d_in[4];
    const float* W3  = (const float*)d_in[5];
    const float* b3  = (const float*)d_in[6];
    const int*   src = (const int*)d_in[7];
    const int*   dst = (const int*)d_in[8];
    const int*   n2g = (const int*)d_in[9];
    const int n_nodes = in_sizes[0] / D;
    const int E       = in_sizes[7];
    float* out = (float*)d_out;

    // Ping-pong scratch: bufA = linear output m, bufB = edge aggregation.
    float* bufA = (float*)d_ws;
    float* bufB = bufA + (size_t)n_nodes * D;

    const int tiles = (n_nodes + 15) / 16;
    const dim3 linGrid((tiles + TILES_PER_BLOCK - 1) / TILES_PER_BLOCK);
    const dim3 linBlock(LIN_BLOCK);

    const size_t feat = (size_t)n_nodes * D;
    const int zeroBlocks    = (int)((feat + 255) / 256);
    const int scatterBlocks = (int)(((size_t)E * 32 + 255) / 256);

    // Layer 1: no tanh on raw input x.
    linear_wmma_kernel<<<linGrid, linBlock, 0, stream>>>(x, W1, b1, bufA, n_nodes, 0);
    zero_kernel<<<zeroBlocks, 256, 0, stream>>>(bufB, feat);
    scatter_add_kernel<<<scatterBlocks, 256, 0, stream>>>(bufA, src, dst, bufB, E);

    // Layer 2: tanh(agg1) fused into A-fragment load.
    linear_wmma_kernel<<<linGrid, linBlock, 0, stream>>>(bufB, W2, b2, bufA, n_nodes, 1);
    zero_kernel<<<zeroBlocks, 256, 0, stream>>>(bufB, feat);
    scatter_add_kernel<<<scatterBlocks, 256, 0, stream>>>(bufA, src, dst, bufB, E);

    // Layer 3: tanh(agg2) fused; no tanh after aggregation.
    linear_wmma_kernel<<<linGrid, linBlock, 0, stream>>>(bufB, W3, b3, bufA, n_nodes, 1);
    zero_kernel<<<zeroBlocks, 256, 0, stream>>>(bufB, feat);
    scatter_add_kernel<<<scatterBlocks, 256, 0, stream>>>(bufA, src, dst, bufB, E);

    // Per-graph scalar reduction into d_out (8 floats, zeroed first).
    zero_kernel<<<1, 32, 0, stream>>>(out, (size_t)out_size);
    graph_reduce_kernel<<<512, 256, 0, stream>>>(bufB, n2g, out, n_nodes);
}